// SpikingBrain_43533788512809
// MI455X (gfx1250) — compile-verified
//
#include <hip/hip_runtime.h>

#define N_NEUR 8192
#define BLK    256
#define KBLK   (N_NEUR / BLK)   // 32 blocks
#define NW     8                // 8 wave32s per block

typedef float v2f  __attribute__((ext_vector_type(2)));
typedef float v8f  __attribute__((ext_vector_type(8)));
typedef unsigned int u32x4 __attribute__((ext_vector_type(4)));
typedef int   i32x4 __attribute__((ext_vector_type(4)));
typedef int   i32x8 __attribute__((ext_vector_type(8)));

// Re-arm cross-workgroup flags every launch (workspace is not re-poisoned
// between graph replays, and the previous run left them set).
__global__ void sb_init_flags(unsigned int* flags) {
    flags[threadIdx.x] = 0u;
}

__global__ __launch_bounds__(256, 1)
void sb_main(const float* __restrict__ ext,
             const float* __restrict__ W,
             const float* __restrict__ thr,
             const float* __restrict__ pot0,
             float* __restrict__ out,
             unsigned int* __restrict__ flags,
             unsigned int* __restrict__ smask) {
    extern __shared__ float smem[];                       // [0,256KB): diagonal tile
    float*        w_diag  = smem;
    unsigned int* dec_lds = (unsigned int*)(smem + BLK * BLK);

    const int k = blockIdx.x;
    const int t = threadIdx.x;
    const int w = t >> 5;        // wave id (wave32)
    const int l = t & 31;        // lane id
    const int n = k * BLK + t;   // this thread's neuron

    // ---- TDM: prefetch diagonal tile W[kB..kB+255, kB..kB+255] into LDS ----
    // Issued by wave 0 only (TDM ignores EXEC; one DMA per wave that issues).
    // Overlaps with the entire off-diagonal phase; waited on with TENSORcnt.
    if (t < 32) {
        unsigned long long ga =
            (unsigned long long)(const void*)(W + (size_t)k * BLK * N_NEUR + (size_t)k * BLK);
        u32x4 g0;
        g0[0] = 1u;                                        // count = 1 valid descriptor
        g0[1] = 0u;                                        // LDS byte addr: base of dynamic LDS
        g0[2] = (unsigned int)(ga & 0xFFFFFFFFull);        // global_addr[31:0]
        g0[3] = (unsigned int)((ga >> 32) & 0x01FFFFFFull) // global_addr[56:32]
              | (2u << 30);                                // type = 2 ("image")
        i32x8 g1;
        g1[0] = (int)(2u << 16);                           // wg_mask=0, data_size=2 (4 bytes)
        g1[1] = (int)(((unsigned)N_NEUR & 0xFFFFu) << 16);               // tensor_dim0 lo16
        g1[2] = (int)((((unsigned)N_NEUR >> 16) & 0xFFFFu)
              | (((unsigned)N_NEUR & 0xFFFFu) << 16));                   // dim0 hi16 | dim1 lo16
        g1[3] = (int)((((unsigned)N_NEUR >> 16) & 0xFFFFu)
              | ((unsigned)BLK << 16));                                  // dim1 hi16 | tile_dim0
        g1[4] = (int)BLK;                                  // tile_dim1 = 256, tile_dim2 = 0 (2D)
        g1[5] = (int)N_NEUR;                               // tensor_dim0_stride lo32 = 8192
        g1[6] = 0;                                         // stride hi | dim1_stride lo (unused 2D)
        g1[7] = 0;
        i32x4 gz4; gz4[0] = 0; gz4[1] = 0; gz4[2] = 0; gz4[3] = 0;
        i32x8 gz8; gz8[0] = 0; gz8[1] = 0; gz8[2] = 0; gz8[3] = 0;
                   gz8[4] = 0; gz8[5] = 0; gz8[6] = 0; gz8[7] = 0;
        // 6-arg form on this toolchain (clang-23 / therock headers).
        __builtin_amdgcn_tensor_load_to_lds(g0, g1, gz4, gz4, gz8, 0);
    }

    float acc = pot0[n] + ext[n];
    const float th = thr[n];

    // ---- Phase A: off-diagonal accumulation via WMMA f32 16x16x4 ----
    // Broadcast trick: A[M][K] = s[jj+K] for every M  =>  every row of D is the
    // 16-wide partial matvec for columns [cb, cb+16). Lane extracts element 0.
    // Two chains cover this wave's 32 columns; a/b accumulator pairs break the
    // WMMA RAW chain. W loads are fully coalesced and 100% useful.
    v8f c0a = {}, c0b = {}, c1a = {}, c1b = {};
    const int cb0 = k * BLK + w * 32;
    const int lh  = (l < 16) ? 0 : 1;
    const int cA  = cb0 + (l & 15);
    const int cB  = cA + 16;

    for (int m = 0; m < k; ++m) {
        if (t == 0) {
            while (__hip_atomic_load(&flags[m], __ATOMIC_ACQUIRE,
                                     __HIP_MEMORY_SCOPE_AGENT) == 0u) {
                __builtin_amdgcn_s_sleep(2);
            }
        }
        __syncthreads();

        unsigned int mw[8];
#pragma unroll
        for (int q = 0; q < 8; ++q)
            mw[q] = __hip_atomic_load(&smask[m * 8 + q], __ATOMIC_RELAXED,
                                      __HIP_MEMORY_SCOPE_AGENT);

        const size_t rbase = (size_t)m * BLK * N_NEUR;
#pragma unroll 4
        for (int jj = 0; jj < BLK; jj += 8) {
            {
                unsigned int word = mw[jj >> 5];
                unsigned int sh   = (unsigned)(jj & 31) + ((l < 16) ? 0u : 2u);
                v2f a; a.x = (float)((word >> sh) & 1u);
                       a.y = (float)((word >> (sh + 1u)) & 1u);
                const size_t r0 = rbase + (size_t)(jj + lh) * N_NEUR;
                const size_t r1 = rbase + (size_t)(jj + 2 + lh) * N_NEUR;
                v2f b0; b0.x = W[r0 + cA]; b0.y = W[r1 + cA];
                v2f b1; b1.x = W[r0 + cB]; b1.y = W[r1 + cB];
                c0a = __builtin_amdgcn_wmma_f32_16x16x4_f32(false, a, false, b0,
                                                            (short)0, c0a, false, false);
                c1a = __builtin_amdgcn_wmma_f32_16x16x4_f32(false, a, false, b1,
                                                            (short)0, c1a, false, false);
            }
            {
                const int j2 = jj + 4;
                unsigned int word = mw[j2 >> 5];
                unsigned int sh   = (unsigned)(j2 & 31) + ((l < 16) ? 0u : 2u);
                v2f a; a.x = (float)((word >> sh) & 1u);
                       a.y = (float)((word >> (sh + 1u)) & 1u);
                const size_t r0 = rbase + (size_t)(j2 + lh) * N_NEUR;
                const size_t r1 = rbase + (size_t)(j2 + 2 + lh) * N_NEUR;
                v2f b0; b0.x = W[r0 + cA]; b0.y = W[r1 + cA];
                v2f b1; b1.x = W[r0 + cB]; b1.y = W[r1 + cB];
                c0b = __builtin_amdgcn_wmma_f32_16x16x4_f32(false, a, false, b0,
                                                            (short)0, c0b, false, false);
                c1b = __builtin_amdgcn_wmma_f32_16x16x4_f32(false, a, false, b1,
                                                            (short)0, c1b, false, false);
            }
        }
    }
    // Lane l owns column cb0+l: halves map to chain0 / chain1 respectively.
    acc += (l < 16) ? (c0a[0] + c0b[0]) : (c1a[0] + c1b[0]);

    // ---- Phase D: sequential diagonal (forward substitution) ----
    if (t < 32) __builtin_amdgcn_s_wait_tensorcnt(0);   // tile arrived
    __syncthreads();

    for (int wp = 0; wp < NW; ++wp) {
        if (w == wp) {
            // 32 barrier-free sequential steps inside the owning wave.
            unsigned int dmask = 0u;
#pragma unroll 8
            for (int i2 = 0; i2 < 32; ++i2) {
                unsigned int bal = __builtin_amdgcn_ballot_w32(acc >= th);
                unsigned int s   = (bal >> i2) & 1u;
                dmask |= (s << i2);
                acc += (float)s * w_diag[(wp * 32 + i2) * BLK + t];
            }
            if (l == 0) dec_lds[wp] = dmask;
        }
        __syncthreads();
        if (w > wp) {
            // Bulk rank-update for not-yet-decided waves (32 fmacs, LDS reads
            // are conflict-free: consecutive addresses per lane).
            const unsigned int dm = dec_lds[wp];
#pragma unroll 8
            for (int i2 = 0; i2 < 32; ++i2) {
                float sf = (float)((dm >> i2) & 1u);
                acc += sf * w_diag[(wp * 32 + i2) * BLK + t];
            }
        }
        __syncthreads();
    }

    // ---- Publish: float spikes to d_out, bitmasks + release flag for consumers ----
    const unsigned int dm = dec_lds[w];
    out[n] = (float)((dm >> l) & 1u);
    if (l == 0) smask[k * 8 + w] = dm;
    __threadfence();
    __syncthreads();
    if (t == 0)
        __hip_atomic_store(&flags[k], 1u, __ATOMIC_RELEASE, __HIP_MEMORY_SCOPE_AGENT);
}

extern "C" void kernel_launch(void* const* d_in, const int* in_sizes, int n_in,
                              void* d_out, int out_size, void* d_ws, size_t ws_size,
                              hipStream_t stream) {
    (void)in_sizes; (void)n_in; (void)out_size; (void)ws_size;
    const float* ext  = (const float*)d_in[0];   // external_input [N]
    const float* W    = (const float*)d_in[1];   // connections    [N,N] row-major
    const float* thr  = (const float*)d_in[2];   // thresholds     [N]
    const float* pot0 = (const float*)d_in[3];   // init_potentials[N]

    unsigned int* flags = (unsigned int*)d_ws;   // 64 words (padded)
    unsigned int* smask = flags + 64;            // K*8 spike bitmask words

    sb_init_flags<<<1, 64, 0, stream>>>(flags);

    const size_t shmem = (size_t)BLK * BLK * sizeof(float) + 64;  // 256KB tile + masks
    sb_main<<<KBLK, BLK, shmem, stream>>>(ext, W, thr, pot0,
                                          (float*)d_out, flags, smask);
}